// TangentProject_15324443312269
// MI455X (gfx1250) — compile-verified
//
#include <hip/hip_runtime.h>

typedef __attribute__((ext_vector_type(2))) float v2f;
typedef __attribute__((ext_vector_type(8))) float v8f;

namespace {
constexpr int    ND     = 256;                 // matrix dimension N
constexpr size_t NN     = (size_t)ND * ND;     // elements per matrix
constexpr int    BIGMOD = 1 << 30;
constexpr int    CH     = 256;                 // matrices per Stage-C chunk
}

// Async direct-to-LDS copy (CDNA5): copies 16 bytes from global address base + byte_off to LDS address lds_off.
// Tracked by ASYNCcnt; GVS addressing (uniform SGPR base + 32-bit lane offset).
__device__ __forceinline__
void async_ld_b128(uint32_t lds_off, const float* base, uint32_t byte_off)
{
  asm volatile("global_load_async_to_lds_b128 %0, %1, %2"
               :: "v"(lds_off), "v"(byte_off), "s"(base)
               : "memory");
}
__device__ __forceinline__ void wait_async0()
{
  asm volatile("s_wait_asynccnt 0x0" ::: "memory");
}

// ---------------------------------------------------------------------------
// Batched f32 GEMM on the matrix cores:  D[bt] = alpha * A[selA] @ B[selB] + beta*I
// selA = ((bt/divA) % modA), selB = ((bt/divB) % modB)  (enables broadcast of S_c)
// grid = (2, 2, batch), block = 256 threads = 8 waves (wave32).
// Each block: 128x128 tile; each wave: 64x32 tile = 4x2 accumulators of 16x16.
// V_WMMA_F32_16X16X4_F32 compute; double-buffered LDS fed by
// GLOBAL_LOAD_ASYNC_TO_LDS_B128 (software pipelined, ASYNCcnt-synchronized).
// ---------------------------------------------------------------------------
__global__ __launch_bounds__(256)
void gemm_wmma_f32(const float* __restrict__ Abase, const float* __restrict__ Bbase,
                   float* __restrict__ Dbase, float alpha, float beta,
                   int divA, int modA, int divB, int modB)
{
  const int bt = blockIdx.z;
  const float* __restrict__ A = Abase + (size_t)((bt / divA) % modA) * NN;
  const float* __restrict__ B = Bbase + (size_t)((bt / divB) % modB) * NN;
  float* __restrict__ D = Dbase + (size_t)bt * NN;

  const int Mb = blockIdx.x * 128;
  const int Nb = blockIdx.y * 128;

  __shared__ float As[2][128][33];   // odd stride: conflict-friendly A-frag reads
  __shared__ float Bs[2][32][132];

  const int tid  = threadIdx.x;
  const int lane = tid & 31;
  const int wave = tid >> 5;
  const int wr   = wave >> 2;      // 0..1  (64-row slab)
  const int wc   = wave & 3;       // 0..3  (32-col slab)
  const int lh   = lane >> 4;      // half-wave: selects K pair
  const int l16  = lane & 15;

  // Issue one 128x32 A-panel + 32x128 B-panel into LDS buffer pb2 (async).
  auto issue_panel = [&](int pb2, int k0) {
#pragma unroll
    for (int l = 0; l < 4; ++l) {
      int lin = tid + l * 256;           // b128 unit 0..1023
      int row = lin >> 3;
      int c4  = (lin & 7) << 2;
      async_ld_b128((uint32_t)(uintptr_t)&As[pb2][row][c4], A,
                    (uint32_t)(((Mb + row) * ND + k0 + c4) * 4));
    }
#pragma unroll
    for (int l = 0; l < 4; ++l) {
      int lin = tid + l * 256;
      int row = lin >> 5;
      int c4  = (lin & 31) << 2;
      async_ld_b128((uint32_t)(uintptr_t)&Bs[pb2][row][c4], B,
                    (uint32_t)(((k0 + row) * ND + Nb + c4) * 4));
    }
  };

  v8f acc[4][2] = {};

  int pb = 0;
  issue_panel(0, 0);
  wait_async0();
  __syncthreads();                       // buffer 0 visible to all waves

  for (int k0 = 0; k0 < ND; k0 += 32) {
    const bool more = (k0 + 32) < ND;
    if (more) issue_panel(pb ^ 1, k0 + 32);   // overlap next panel with compute

#pragma unroll
    for (int kk = 0; kk < 32; kk += 4) {
      const int kb = kk + lh * 2;        // lanes 0-15: K=kk,kk+1 ; 16-31: K=kk+2,kk+3
      v2f af[4], bf[2];
#pragma unroll
      for (int i = 0; i < 4; ++i) {
        int m = wr * 64 + i * 16 + l16;
        af[i][0] = As[pb][m][kb];
        af[i][1] = As[pb][m][kb + 1];
      }
#pragma unroll
      for (int j = 0; j < 2; ++j) {
        int n = wc * 32 + j * 16 + l16;
        bf[j][0] = Bs[pb][kb][n];
        bf[j][1] = Bs[pb][kb + 1][n];
      }
#pragma unroll
      for (int i = 0; i < 4; ++i)
#pragma unroll
        for (int j = 0; j < 2; ++j)
          acc[i][j] = __builtin_amdgcn_wmma_f32_16x16x4_f32(
              false, af[i], false, bf[j], (short)0, acc[i][j], false, false);
    }

    if (more) {
      wait_async0();                     // my next-panel async loads done
      __syncthreads();                   // everyone done reading pb + loads visible
      pb ^= 1;
    }
  }

  // Epilogue: D = alpha*acc + beta*I. C/D layout: VGPR v -> row v (+8 upper half), col l16.
#pragma unroll
  for (int i = 0; i < 4; ++i)
#pragma unroll
    for (int j = 0; j < 2; ++j)
#pragma unroll
      for (int v = 0; v < 8; ++v) {
        int m = Mb + wr * 64 + i * 16 + v + lh * 8;
        int n = Nb + wc * 32 + j * 16 + l16;
        D[(size_t)m * ND + n] = alpha * acc[i][j][v] + ((m == n) ? beta : 0.0f);
      }
}

// ---------------------------------------------------------------------------
// Per-matrix infinity norm (Gershgorin bound on spectral radius).
// ---------------------------------------------------------------------------
__global__ __launch_bounds__(256)
void rownorm_inf(const float* __restrict__ A, float* __restrict__ norms)
{
  const int bt = blockIdx.x;
  const float* M = A + (size_t)bt * NN;
  const int i = threadIdx.x;
  float s = 0.0f;
  for (int j = 0; j < ND; j += 4) {
    float4 v = *(const float4*)(M + (size_t)i * ND + j);
    s += fabsf(v.x) + fabsf(v.y) + fabsf(v.z) + fabsf(v.w);
  }
  __shared__ float red[256];
  red[i] = s;
  __syncthreads();
  for (int off = 128; off > 0; off >>= 1) {
    if (i < off) red[i] = fmaxf(red[i], red[i + off]);
    __syncthreads();
  }
  if (i == 0) norms[bt] = red[0];
}

// D = cA*A + cI*I, optionally scaled by 1/norm (mode 1) or rsqrt(norm) (mode 2).
__global__ __launch_bounds__(256)
void ew_affine(const float* __restrict__ A, float* __restrict__ D,
               float cA, float cI, const float* __restrict__ norms, int mode,
               size_t total)
{
  size_t idx = (size_t)blockIdx.x * 256 + threadIdx.x;
  if (idx >= total) return;
  size_t bt = idx >> 16;
  int r = (int)(idx & 65535);
  int i = r >> 8, j = r & 255;
  float v = cA * A[idx] + ((i == j) ? cI : 0.0f);
  if (mode == 1)       v *= 1.0f / norms[bt];
  else if (mode == 2)  v *= rsqrtf(norms[bt]);
  D[idx] = v;
}

__global__ __launch_bounds__(256)
void ew_set_identity(float* __restrict__ D, size_t total)
{
  size_t idx = (size_t)blockIdx.x * 256 + threadIdx.x;
  if (idx >= total) return;
  int r = (int)(idx & 65535);
  D[idx] = ((r >> 8) == (r & 255)) ? 1.0f : 0.0f;
}

// OUT = scale*L + ln(norm)*I   (undo the log-normalization)
__global__ __launch_bounds__(256)
void ew_final_log(const float* __restrict__ L, float* __restrict__ D,
                  const float* __restrict__ norms, float scale, size_t total)
{
  size_t idx = (size_t)blockIdx.x * 256 + threadIdx.x;
  if (idx >= total) return;
  size_t bt = idx >> 16;
  int r = (int)(idx & 65535);
  int i = r >> 8, j = r & 255;
  D[idx] = scale * L[idx] + ((i == j) ? logf(norms[bt]) : 0.0f);
}

// ---------------------------------------------------------------------------
// Host orchestration
// ---------------------------------------------------------------------------
extern "C" void kernel_launch(void* const* d_in, const int* in_sizes, int n_in,
                              void* d_out, int out_size, void* d_ws, size_t ws_size,
                              hipStream_t stream)
{
  (void)in_sizes; (void)n_in; (void)out_size; (void)ws_size;
  const float* X = (const float*)d_in[0];   // input  (256, 256, 256)
  const float* W = (const float*)d_in[1];   // weight (4, 256, 256)
  float* out = (float*)d_out;               // (256, 4, 256, 256)

  // Workspace layout (~337 MB): S (4 mats) | norms | 5 chunk buffers of 256 mats.
  float* base   = (float*)d_ws;
  float* S      = base;                         // 4*NN
  float* normsA = base + 4 * NN;                // 4
  float* normsC = base + 4 * NN + 16;           // 256
  const size_t BUF = (size_t)CH * NN;           // 64 MB each
  float* Bf[5];
  for (int i = 0; i < 5; ++i) Bf[i] = base + 4 * NN + 1024 + (size_t)i * BUF;

  auto sw = [](float*& a, float*& b) { float* t = a; a = b; b = t; };

  // ========== Stage A: S_c = (0.95 W + 0.05 I)^(-1/2) via coupled Newton-Schulz
  {
    const size_t tot = 4 * NN;
    const unsigned eb = (unsigned)(tot / 256);
    ew_affine<<<eb, 256, 0, stream>>>(W, Bf[0], 0.95f, 0.05f, nullptr, 0, tot);
    rownorm_inf<<<4, 256, 0, stream>>>(Bf[0], normsA);
    ew_affine<<<eb, 256, 0, stream>>>(Bf[0], Bf[0], 1.0f, 0.0f, normsA, 1, tot);   // Y0 = W'/s
    ew_set_identity<<<eb, 256, 0, stream>>>(Bf[1], tot);                            // Z0 = I
    float *Yc = Bf[0], *Zc = Bf[1], *T = Bf[2], *Ya = Bf[3], *Za = Bf[4];
    dim3 g(2, 2, 4);
    for (int it = 0; it < 25; ++it) {
      gemm_wmma_f32<<<g, 256, 0, stream>>>(Zc, Yc, T,  -0.5f, 1.5f, 1, BIGMOD, 1, BIGMOD);
      gemm_wmma_f32<<<g, 256, 0, stream>>>(Yc, T,  Ya,  1.0f, 0.0f, 1, BIGMOD, 1, BIGMOD);
      gemm_wmma_f32<<<g, 256, 0, stream>>>(T,  Zc, Za,  1.0f, 0.0f, 1, BIGMOD, 1, BIGMOD);
      sw(Yc, Ya); sw(Zc, Za);
    }
    // Z -> (W'/s)^(-1/2)  =>  S = Z * s^(-1/2)
    ew_affine<<<eb, 256, 0, stream>>>(Zc, S, 1.0f, 0.0f, normsA, 2, tot);
  }

  // Mercator coefficients c_m = (-1)^(m+1)/m for log(I+E)
  const float cs[9] = {0.f, 1.f, -0.5f, 1.f / 3.f, -0.25f, 0.2f,
                       -1.f / 6.f, 1.f / 7.f, -0.125f};

  // ========== Stages B+C, 4 chunks of 256 (b,c) matrices each
  for (int ck = 0; ck < 4; ++ck) {
    const size_t tot = (size_t)CH * NN;
    const unsigned eb = (unsigned)(tot / 256);
    float* Dch = out + (size_t)ck * CH * NN;
    dim3 g(2, 2, CH);

    // T1[t] = S[t%4] @ X[ck*64 + t/4] ;  M[t] = T1[t] @ S[t%4]
    gemm_wmma_f32<<<g, 256, 0, stream>>>(S, X + (size_t)ck * 64 * NN, Bf[0],
                                         1.0f, 0.0f, 1, 4, 4, BIGMOD);
    gemm_wmma_f32<<<g, 256, 0, stream>>>(Bf[0], S, Bf[1],
                                         1.0f, 0.0f, 1, BIGMOD, 1, 4);

    // A0 = (0.95 M + 0.05 I) / s
    ew_affine<<<eb, 256, 0, stream>>>(Bf[1], Bf[1], 0.95f, 0.05f, nullptr, 0, tot);
    rownorm_inf<<<CH, 256, 0, stream>>>(Bf[1], normsC);
    ew_affine<<<eb, 256, 0, stream>>>(Bf[1], Bf[1], 1.0f, 0.0f, normsC, 1, tot);

    // 5 levels of matrix sqrt via coupled Newton-Schulz (14 iters each)
    float *Yc = Bf[1], *Zc = Bf[2], *T = Bf[0], *Ya = Bf[3], *Za = Bf[4];
    for (int lvl = 0; lvl < 5; ++lvl) {
      ew_set_identity<<<eb, 256, 0, stream>>>(Zc, tot);
      for (int it = 0; it < 14; ++it) {
        gemm_wmma_f32<<<g, 256, 0, stream>>>(Zc, Yc, T,  -0.5f, 1.5f, 1, BIGMOD, 1, BIGMOD);
        gemm_wmma_f32<<<g, 256, 0, stream>>>(Yc, T,  Ya,  1.0f, 0.0f, 1, BIGMOD, 1, BIGMOD);
        gemm_wmma_f32<<<g, 256, 0, stream>>>(T,  Zc, Za,  1.0f, 0.0f, 1, BIGMOD, 1, BIGMOD);
        sw(Yc, Ya); sw(Zc, Za);
      }
    }

    // E = A^(1/32) - I ; Horner: P7 = c8*E + c7*I ; P_{m-1} = E@P_m + c_{m-1}*I
    ew_affine<<<eb, 256, 0, stream>>>(Yc, Zc, 1.0f, -1.0f, nullptr, 0, tot);
    float* E = Zc;
    ew_affine<<<eb, 256, 0, stream>>>(E, Ya, cs[8], cs[7], nullptr, 0, tot);
    float* P = Ya; float* Pn = Za;
    for (int m = 6; m >= 1; --m) {
      gemm_wmma_f32<<<g, 256, 0, stream>>>(E, P, Pn, 1.0f, cs[m], 1, BIGMOD, 1, BIGMOD);
      sw(P, Pn);
    }
    gemm_wmma_f32<<<g, 256, 0, stream>>>(E, P, Pn, 1.0f, 0.0f, 1, BIGMOD, 1, BIGMOD);

    // OUT = 2^5 * log(A^(1/32)) + ln(s)*I
    ew_final_log<<<eb, 256, 0, stream>>>(Pn, Dch, normsC, 32.0f, tot);
  }
}